// net_test_35261681500361
// MI455X (gfx1250) — compile-verified
//
#include <hip/hip_runtime.h>

#define D 128
#define NC 16

typedef __attribute__((ext_vector_type(2))) float v2f;
typedef __attribute__((ext_vector_type(8))) float v8f;

// ---------------------------------------------------------------------------
// SPMM: out[row,:] += val * x[col,:]. One wave (32 lanes) per edge; each lane
// handles 4 contiguous floats via a float4 load + 4 f32 atomics. The 51.2 MB
// feature matrix is resident in the 192 MB L2, so gathers and atomics resolve
// in L2 rather than HBM — this phase is L2-atomic-throughput bound.
// ---------------------------------------------------------------------------
__global__ void spmm_kernel(const float* __restrict__ x,
                            const int* __restrict__ erow,
                            const int* __restrict__ ecol,
                            const float* __restrict__ eval,
                            float* __restrict__ out, int n_edges) {
    int tid  = blockIdx.x * blockDim.x + threadIdx.x;
    int edge = tid >> 5;
    int lane = tid & 31;
    if (edge >= n_edges) return;
    int   r = erow[edge];
    int   c = ecol[edge];
    float v = eval[edge];
    const float4 m = ((const float4*)(x + (size_t)c * D))[lane];
    float* dst = out + (size_t)r * D + lane * 4;
    atomicAdd(dst + 0, v * m.x);
    atomicAdd(dst + 1, v * m.y);
    atomicAdd(dst + 2, v * m.z);
    atomicAdd(dst + 3, v * m.w);
}

// ---------------------------------------------------------------------------
// Dense GEMM + ReLU via V_WMMA_F32_16X16X4_F32 (fp32-precise matrix path).
// W is staged into LDS ONCE per 8-wave block, pre-swizzled into K-pair
// float2s: wswz[p*128 + n] = {W[2p][n], W[2p+1][n]}. That makes each WMMA B
// fragment a single aligned ds_load_b64 into the even-aligned VGPR pair the
// instruction consumes — no register shuffles in the hot loop. A fragments
// stream from global (VMEM) while B comes from LDS (DS): both pipes overlap.
// ---------------------------------------------------------------------------
__global__ void __launch_bounds__(256)
gemm_relu_wmma(const float* __restrict__ A,   // [n x 128]
               const float* __restrict__ W,   // [128 x 128]
               float* __restrict__ O,         // [n x 128]
               int n_strips) {
    __shared__ v2f wswz[(D / 2) * D];         // 64 KB of the 320 KB/WGP LDS

    // --- swizzled staging: quad q covers pair-row p, columns n4..n4+3 ---
    for (int q = threadIdx.x; q < (D / 2) * D / 4; q += 256) {
        const int p  = q >> 5;                // q / (D/4)
        const int n4 = (q & 31) * 4;
        const float4 r0 = *(const float4*)(W + (size_t)(2 * p)     * D + n4);
        const float4 r1 = *(const float4*)(W + (size_t)(2 * p + 1) * D + n4);
        v2f* dst = &wswz[p * D + n4];
        v2f t0; t0.x = r0.x; t0.y = r1.x; dst[0] = t0;
        v2f t1; t1.x = r0.y; t1.y = r1.y; dst[1] = t1;
        v2f t2; t2.x = r0.z; t2.y = r1.z; dst[2] = t2;
        v2f t3; t3.x = r0.w; t3.y = r1.w; dst[3] = t3;
    }
    __syncthreads();

    int wave = (int)((blockIdx.x * blockDim.x + threadIdx.x) >> 5);
    int lane = threadIdx.x & 31;
    if (wave >= n_strips) wave = n_strips - 1;   // wave-uniform clamp: EXEC stays all-1s

    const int n16 = lane & 15;   // N (and M-row for the A fragment)
    const int kh  = lane >> 4;   // half-wave selector

    const float* arow = A + ((size_t)wave * 16 + n16) * D;

    v8f acc[8] = {};
    for (int k = 0; k < D; k += 4) {
        const int ka = k + kh * 2;
        const int p  = (k >> 1) + kh;         // K-pair index into swizzled W
        v2f a;
        a.x = arow[ka];
        a.y = arow[ka + 1];
#pragma unroll
        for (int t = 0; t < 8; ++t) {
            const v2f b = wswz[p * D + t * 16 + n16];   // single ds_load_b64
            acc[t] = __builtin_amdgcn_wmma_f32_16x16x4_f32(
                false, a, false, b, (short)0, acc[t], false, false);
        }
    }
#pragma unroll
    for (int t = 0; t < 8; ++t) {
#pragma unroll
        for (int v = 0; v < 8; ++v) {
            const int M = kh * 8 + v;
            float f = acc[t][v];
            O[((size_t)wave * 16 + M) * D + t * 16 + n16] = f > 0.f ? f : 0.f;
        }
    }
}

// ---------------------------------------------------------------------------
// Classifier: logits = H[n x 128] @ CLS[128 x 16], hard threshold (>0 -> 1).
// N_CLASSES == 16 -> exactly one WMMA tile per 16-row strip. CLS (8 KB) gets
// the same K-pair swizzle into LDS so B fragments are single ds_load_b64s.
// ---------------------------------------------------------------------------
__global__ void __launch_bounds__(128)
classifier_wmma(const float* __restrict__ A,   // [n x 128]
                const float* __restrict__ WC,  // [128 x 16]
                float* __restrict__ O,         // [n x 16]
                int n_strips) {
    __shared__ v2f cswz[(D / 2) * NC];         // 8 KB

    for (int i = threadIdx.x; i < (D / 2) * NC; i += 128) {
        const int p = i >> 4;
        const int n = i & 15;
        v2f t;
        t.x = WC[(size_t)(2 * p)     * NC + n];
        t.y = WC[(size_t)(2 * p + 1) * NC + n];
        cswz[i] = t;
    }
    __syncthreads();

    int wave = (int)((blockIdx.x * blockDim.x + threadIdx.x) >> 5);
    int lane = threadIdx.x & 31;
    if (wave >= n_strips) wave = n_strips - 1;

    const int n16 = lane & 15;
    const int kh  = lane >> 4;

    const float* arow = A + ((size_t)wave * 16 + n16) * D;

    v8f acc = {};
    for (int k = 0; k < D; k += 4) {
        const int ka = k + kh * 2;
        const int p  = (k >> 1) + kh;
        v2f a;
        a.x = arow[ka];
        a.y = arow[ka + 1];
        const v2f b = cswz[p * NC + n16];
        acc = __builtin_amdgcn_wmma_f32_16x16x4_f32(
            false, a, false, b, (short)0, acc, false, false);
    }
#pragma unroll
    for (int v = 0; v < 8; ++v) {
        const int M = kh * 8 + v;
        O[((size_t)wave * 16 + M) * NC + n16] = acc[v] > 0.f ? 1.f : 0.f;
    }
}

extern "C" void kernel_launch(void* const* d_in, const int* in_sizes, int n_in,
                              void* d_out, int out_size, void* d_ws, size_t ws_size,
                              hipStream_t stream) {
    const float* x    = (const float*)d_in[0];
    const int*   erow = (const int*)  d_in[1];
    const int*   ecol = (const int*)  d_in[2];
    const float* eval = (const float*)d_in[3];
    const float* w1   = (const float*)d_in[4];
    const float* w2   = (const float*)d_in[5];
    const float* cls  = (const float*)d_in[6];
    float* out = (float*)d_out;

    const int n_nodes  = in_sizes[0] / D;     // 100000 (= 6250 * 16)
    const int n_edges  = in_sizes[1];         // 1.6M
    const int n_strips = n_nodes / 16;        // 6250

    const size_t feat_bytes = (size_t)n_nodes * D * sizeof(float);
    float* bufA = (float*)d_ws;                                // spmm accumulator
    float* bufB = (float*)((char*)d_ws + feat_bytes);          // post-GEMM features

    const int spmm_block = 256;                                // 8 edges / block
    const int spmm_grid  = (n_edges + 7) / 8;
    const int gemm_block = 256;                                // 8 waves / block
    const int gemm_grid  = (n_strips + 7) / 8;
    const int cls_block  = 128;                                // 4 waves / block
    const int cls_grid   = (n_strips + 3) / 4;

    // Layer 1: s1 = spmm(x); h1 = relu(s1 @ w1)
    hipMemsetAsync(bufA, 0, feat_bytes, stream);
    spmm_kernel<<<spmm_grid, spmm_block, 0, stream>>>(x, erow, ecol, eval, bufA, n_edges);
    gemm_relu_wmma<<<gemm_grid, gemm_block, 0, stream>>>(bufA, w1, bufB, n_strips);

    // Layer 2: s2 = spmm(h1); h2 = relu(s2 @ w2)
    hipMemsetAsync(bufA, 0, feat_bytes, stream);
    spmm_kernel<<<spmm_grid, spmm_block, 0, stream>>>(bufB, erow, ecol, eval, bufA, n_edges);
    gemm_relu_wmma<<<gemm_grid, gemm_block, 0, stream>>>(bufA, w2, bufB, n_strips);

    // Classifier + hard threshold
    classifier_wmma<<<cls_grid, cls_block, 0, stream>>>(bufB, cls, out, n_strips);
}